// RFP_DAGs_50929722196486
// MI455X (gfx1250) — compile-verified
//
#include <hip/hip_runtime.h>

#define Cc 64
#define Hh 64
#define Ww 64
#define Bb 4
#define Dd 8
#define Gg 4
#define ST 68   // padded LDS row stride (floats): conflict-free half-wave banks, 8B aligned

typedef float v2f __attribute__((ext_vector_type(2)));
typedef float v8f __attribute__((ext_vector_type(8)));

// ---------------------------------------------------------------------------
// Kernel 1: x [B,C,H,W,D] -> xt [D,B,H,W,C]  (channels-last, slice-major)
// ---------------------------------------------------------------------------
__global__ __launch_bounds__(256, 2)
void xpose_kernel(const float* __restrict__ x, float* __restrict__ xt)
{
    __shared__ float tile[64][130];          // [c][(w-w0)*8+d]
    const int w0 = blockIdx.x * 16;
    const int h  = blockIdx.y;
    const int b  = blockIdx.z;

    const float* src = x + (((size_t)b * Cc) * Hh + h) * (Ww * Dd) + (size_t)w0 * Dd;
    const int l128 = threadIdx.x & 127;
    for (int c = (threadIdx.x >> 7); c < Cc; c += 2)
        tile[c][l128] = src[(size_t)c * Hh * Ww * Dd + l128];
    __syncthreads();

    const int c   = threadIdx.x & 63;
    const int wd0 = threadIdx.x >> 6;        // 0..3
    for (int q = wd0; q < 128; q += 4) {
        const int w = w0 + (q >> 3);
        const int d = q & 7;
        xt[(((size_t)(d * Bb + b) * Hh + h) * Ww + w) * Cc + c] = tile[c][q];
    }
}

// ---------------------------------------------------------------------------
// Kernel 2: per-(g,d,b) anti-diagonal DAG scan with fp32 WMMA.
//   512 threads = 16 waves; one 16x16 D-tile per wave; weight (B) fragments
//   resident in VGPRs for the whole scan; two interleaved accumulators.
// ---------------------------------------------------------------------------
__global__ __launch_bounds__(512, 1)
void scan_kernel(const float* __restrict__ xt,
                 const float* __restrict__ gWx,
                 const float* __restrict__ gWv,
                 const float* __restrict__ gWh,
                 const float* __restrict__ gB,
                 float* __restrict__ hbuf)
{
    extern __shared__ float smem[];
    float* sWx = smem;                       // [64][ST]  W[o][c]
    float* sWv = sWx + 64 * ST;
    float* sWh = sWv + 64 * ST;
    float* sX  = sWh + 64 * ST;              // [64][ST]  X[cell][c]
    float* hA  = sX  + 64 * ST;              // [2][65][ST]; row0 = zeros, cell i at row i+1
    float* sB  = hA  + 2 * 65 * ST;          // [64]

    const int wg = blockIdx.x;               // 0..127
    const int b  = wg & 3;
    const int d  = (wg >> 2) & 7;
    const int g  = wg >> 5;
    const int fh = (g >> 1) & 1;             // _FLIPS = (F,F),(F,T),(T,F),(T,T)
    const int fw = g & 1;

    // stage weights + bias, zero h diagonals
    {
        const size_t woff = (size_t)(g * Dd + d) * Cc * Cc;
        const float* wx = gWx + woff;
        const float* wv = gWv + woff;
        const float* wh = gWh + woff;
        for (int q = threadIdx.x; q < Cc * Cc; q += 512) {
            const int o = q >> 6, c = q & 63;
            sWx[o * ST + c] = wx[q];
            sWv[o * ST + c] = wv[q];
            sWh[o * ST + c] = wh[q];
        }
        for (int q = threadIdx.x; q < 2 * 65 * ST; q += 512) hA[q] = 0.0f;
        if (threadIdx.x < 64) sB[threadIdx.x] = gB[(g * Dd + d) * Cc + threadIdx.x];
    }
    __syncthreads();

    const int lane = threadIdx.x & 31;
    const int wvid = threadIdx.x >> 5;       // wave 0..15
    const int l15  = lane & 15;
    const int kk   = (lane >> 4) << 1;       // 0 or 2
    const int mT   = (wvid & 3) << 4;        // 0,16,32,48
    const int nT   = (wvid >> 2) << 4;       // 0,16,32,48

    // ---- hoist B = W^T fragments into registers for the whole scan ----
    v2f bx[16], bv[16], bh[16];
    {
        const float* WxR = sWx + (nT + l15) * ST + kk;
        const float* WvR = sWv + (nT + l15) * ST + kk;
        const float* WhR = sWh + (nT + l15) * ST + kk;
        #pragma unroll
        for (int s = 0; s < 16; ++s) {
            bx[s] = (v2f){WxR[4 * s], WxR[4 * s + 1]};
            bv[s] = (v2f){WvR[4 * s], WvR[4 * s + 1]};
            bh[s] = (v2f){WhR[4 * s], WhR[4 * s + 1]};
        }
    }
    const float bias = sB[nT + l15];

    const float* xplane = xt   + (size_t)(d * Bb + b) * Hh * Ww * Cc;
    float*       hplane = hbuf + (size_t)((g * Dd + d) * Bb + b) * Hh * Ww * Cc;

    int pp = 0;                              // parity: prev buffer
    for (int t = 0; t < Hh + Ww - 1; ++t) {
        const int lo = (t - (Ww - 1)) > 0 ? (t - (Ww - 1)) : 0;
        const int hi = t < (Hh - 1) ? t : (Hh - 1);

        // stage X rows for this diagonal (coalesced 256B per cell)
        for (int q = threadIdx.x; q < 64 * 16; q += 512) {
            const int i = q >> 4, f4 = q & 15;
            if (i >= lo && i <= hi) {
                const int j  = t - i;
                const int r  = fh ? (Hh - 1 - i) : i;
                const int cw = fw ? (Ww - 1 - j) : j;
                const float4 v = ((const float4*)(xplane + ((size_t)r * Ww + cw) * Cc))[f4];
                *(float4*)(sX + i * ST + f4 * 4) = v;
            }
        }
        __syncthreads();

        const float* prev = hA + pp * 65 * ST;
        float*       cur  = hA + (1 - pp) * 65 * ST;

        v8f acc0 = {bias, bias, bias, bias, bias, bias, bias, bias};
        v8f acc1 = {0.f, 0.f, 0.f, 0.f, 0.f, 0.f, 0.f, 0.f};

        const float* Xrow  = sX   + (mT + l15) * ST + kk;
        const float* Uprow = prev + (mT + l15) * ST + kk;   // cell m-1 (row m; row0=0)
        const float* Lfrow = Uprow + ST;                    // cell m   (row m+1)

        #pragma unroll
        for (int s = 0; s < 16; s += 2) {
            v2f a0 = (v2f){Xrow[4 * s],     Xrow[4 * s + 1]};
            v2f a1 = (v2f){Xrow[4 * s + 4], Xrow[4 * s + 5]};
            acc0 = __builtin_amdgcn_wmma_f32_16x16x4_f32(
                       false, a0, false, bx[s],     (short)0, acc0, false, false);
            acc1 = __builtin_amdgcn_wmma_f32_16x16x4_f32(
                       false, a1, false, bx[s + 1], (short)0, acc1, false, false);
        }
        #pragma unroll
        for (int s = 0; s < 16; s += 2) {
            v2f a0 = (v2f){Uprow[4 * s],     Uprow[4 * s + 1]};
            v2f a1 = (v2f){Uprow[4 * s + 4], Uprow[4 * s + 5]};
            acc0 = __builtin_amdgcn_wmma_f32_16x16x4_f32(
                       false, a0, false, bv[s],     (short)0, acc0, false, false);
            acc1 = __builtin_amdgcn_wmma_f32_16x16x4_f32(
                       false, a1, false, bv[s + 1], (short)0, acc1, false, false);
        }
        #pragma unroll
        for (int s = 0; s < 16; s += 2) {
            v2f a0 = (v2f){Lfrow[4 * s],     Lfrow[4 * s + 1]};
            v2f a1 = (v2f){Lfrow[4 * s + 4], Lfrow[4 * s + 5]};
            acc0 = __builtin_amdgcn_wmma_f32_16x16x4_f32(
                       false, a0, false, bh[s],     (short)0, acc0, false, false);
            acc1 = __builtin_amdgcn_wmma_f32_16x16x4_f32(
                       false, a1, false, bh[s + 1], (short)0, acc1, false, false);
        }

        // relu + masked store of the new diagonal into LDS
        #pragma unroll
        for (int r = 0; r < 8; ++r) {
            const int M    = r + ((lane >> 4) << 3);
            const int cell = mT + M;
            float v = acc0[r] + acc1[r];
            v = v > 0.0f ? v : 0.0f;
            if (cell < lo || cell > hi) v = 0.0f;
            cur[(cell + 1) * ST + nT + l15] = v;
        }
        __syncthreads();

        // write the valid cells of this diagonal to global (un-flipped coords)
        for (int q = threadIdx.x; q < 64 * 16; q += 512) {
            const int i = q >> 4, f4 = q & 15;
            if (i >= lo && i <= hi) {
                const int j  = t - i;
                const int r  = fh ? (Hh - 1 - i) : i;
                const int cw = fw ? (Ww - 1 - j) : j;
                const float4 v = *(const float4*)(cur + (i + 1) * ST + f4 * 4);
                ((float4*)(hplane + ((size_t)r * Ww + cw) * Cc))[f4] = v;
            }
        }
        pp ^= 1;
    }
}

// ---------------------------------------------------------------------------
// Kernel 3: out[b,c,h,w,d] = relu(x + 0.25 * sum_g hbuf[g,d,b,h,w,c])
// ---------------------------------------------------------------------------
__global__ __launch_bounds__(256, 2)
void combine_kernel(const float* __restrict__ xt,
                    const float* __restrict__ hbuf,
                    float* __restrict__ out)
{
    __shared__ float s[Dd][Cc + 1];
    const size_t p = blockIdx.x;             // 0..B*H*W-1
    const int w = p & 63;
    const int h = (p >> 6) & 63;
    const int b = (int)(p >> 12);

    const size_t gstride = (size_t)Dd * Bb * Hh * Ww * Cc;
    for (int it = 0; it < 2; ++it) {
        const int idx = it * 256 + threadIdx.x;
        const int d = idx >> 6, c = idx & 63;
        const size_t base = (((size_t)(d * Bb + b) * Hh + h) * Ww + w) * Cc + c;
        float acc = hbuf[base];
        acc += hbuf[gstride + base];
        acc += hbuf[2 * gstride + base];
        acc += hbuf[3 * gstride + base];
        float v = xt[base] + 0.25f * acc;
        s[d][c] = v > 0.0f ? v : 0.0f;
    }
    __syncthreads();
    for (int it = 0; it < 2; ++it) {
        const int idx = it * 256 + threadIdx.x;
        const int d = idx & 7, c = idx >> 3;
        out[(((size_t)b * Cc + c) * Hh + h) * (Ww * Dd) + (size_t)w * Dd + d] = s[d][c];
    }
}

// ---------------------------------------------------------------------------
extern "C" void kernel_launch(void* const* d_in, const int* in_sizes, int n_in,
                              void* d_out, int out_size, void* d_ws, size_t ws_size,
                              hipStream_t stream)
{
    const float* x  = (const float*)d_in[0];   // [B,C,H,W,D]
    const float* Wx = (const float*)d_in[1];   // [G,D,C,C]
    const float* Wv = (const float*)d_in[2];
    const float* Wh = (const float*)d_in[3];
    const float* bi = (const float*)d_in[4];   // [G,D,C]
    float* out = (float*)d_out;

    const size_t nXT  = (size_t)Dd * Bb * Hh * Ww * Cc;      // 8.39M floats
    float* xt   = (float*)d_ws;                              // [D,B,H,W,C]
    float* hbuf = xt + nXT;                                  // [G,D,B,H,W,C]

    const size_t scan_smem =
        (3 * 64 * ST + 64 * ST + 2 * 65 * ST + 64) * sizeof(float);   // ~103 KB
    hipFuncSetAttribute((const void*)scan_kernel,
                        hipFuncAttributeMaxDynamicSharedMemorySize,
                        (int)scan_smem);

    xpose_kernel<<<dim3(Ww / 16, Hh, Bb), 256, 0, stream>>>(x, xt);
    scan_kernel<<<dim3(Gg * Dd * Bb), 512, scan_smem, stream>>>(xt, Wx, Wv, Wh, bi, hbuf);
    combine_kernel<<<dim3(Bb * Hh * Ww), 256, 0, stream>>>(xt, hbuf, out);
}